// GINEncoderLayerConcat_70806830841988
// MI455X (gfx1250) — compile-verified
//
#include <hip/hip_runtime.h>
#include <hip/hip_bf16.h>

typedef __attribute__((ext_vector_type(2))) float v2f;
typedef __attribute__((ext_vector_type(8))) float v8f;

__device__ __forceinline__ v8f wmma_f32_16x16x4(v2f a, v2f b, v8f c) {
    // 8 args: (neg_a, A, neg_b, B, c_mod, C, reuse_a, reuse_b)
    return __builtin_amdgcn_wmma_f32_16x16x4_f32(false, a, false, b, (short)0, c,
                                                 false, false);
}

// ---------------------------------------------------------------------------
// Edge scatter-add: agg[dst] += h[src]   (d = 2, thread per edge)
// ---------------------------------------------------------------------------
__global__ __launch_bounds__(256) void gin_scatter_d2(
    const float* __restrict__ h, const int* __restrict__ src,
    const int* __restrict__ dst, float* __restrict__ agg, int n_edges)
{
    int e = blockIdx.x * blockDim.x + threadIdx.x;
    if (e < n_edges) {
        int s = src[e], d = dst[e];
        atomicAdd(&agg[d * 2 + 0], h[s * 2 + 0]);
        atomicAdd(&agg[d * 2 + 1], h[s * 2 + 1]);
    }
}

// d = 64: 64 threads per edge; edge index is wave-uniform -> scalar src/dst
__global__ __launch_bounds__(256) void gin_scatter_d64(
    const float* __restrict__ h, const int* __restrict__ src,
    const int* __restrict__ dst, float* __restrict__ agg, int n_edges)
{
    long long t = (long long)blockIdx.x * blockDim.x + threadIdx.x;
    int e = __builtin_amdgcn_readfirstlane((int)(t >> 6));  // uniform per wave32
    int f = (int)(t & 63);
    if (e < n_edges) {
        int s = __builtin_amdgcn_readfirstlane(src[e]);
        int d = __builtin_amdgcn_readfirstlane(dst[e]);
        const float* __restrict__ hrow = h + (size_t)s * 64;
        float*       __restrict__ arow = agg + (size_t)d * 64;
        atomicAdd(&arow[f], hrow[f]);
    }
}

// ---------------------------------------------------------------------------
// MLP: h_out = relu( relu((h_in+agg) @ W1 + b1) @ W2 + b2 )
// One wave32 per 16-node tile. fp32 WMMA 16x16x4 accumulation.
// A layout (16x4 MxK, 32-bit): lanes 0-15: {K=k0,k0+1}; lanes 16-31: {K=k0+2,k0+3}
// B layout (4x16 KxN): mirrored; C/D: lane = N, VGPR v = row v + 8*(lane>=16)
// Out-of-range tile rows: A-fragment addresses are CLAMPED (valid, finite data);
// garbage in A row r only affects D row r, whose store is guarded. No exec
// manipulation inside the WMMA loop.
// ---------------------------------------------------------------------------
__global__ __launch_bounds__(32) void gin_mlp_wmma(
    const float* __restrict__ h_in, const float* __restrict__ agg,
    const float* __restrict__ W1, const float* __restrict__ b1,
    const float* __restrict__ W2, const float* __restrict__ b2,
    float* __restrict__ h_out, int n_nodes, int d_in)
{
    __shared__ float t_lds[16 * 64];   // intermediate activation, 4 KB

    const int lane  = threadIdx.x;        // 0..31
    const int r     = lane & 15;          // row within tile (A) / column (B,C)
    const int hh    = lane >> 4;          // half-wave: 0 or 1
    const int node0 = blockIdx.x * 16;

    int rowc = node0 + r;
    if (rowc > n_nodes - 1) rowc = n_nodes - 1;   // clamp (always valid address)

    // -------------------- matmul 1: z @ W1 --------------------
    v8f acc[4] = {v8f{}, v8f{}, v8f{}, v8f{}};

    if (d_in == 2) {
        // K = 2, padded to 4: B upper half is zero, so A upper half is don't-care
        // (but must be finite) -> load from a clamped/safe address and select.
        int base = (hh == 0) ? (rowc * 2) : 0;
        float zx = h_in[base + 0] + agg[base + 0];
        float zy = h_in[base + 1] + agg[base + 1];
        v2f a;
        a.x = (hh == 0) ? zx : 0.0f;
        a.y = (hh == 0) ? zy : 0.0f;
        for (int nt = 0; nt < 4; ++nt) {
            int col = nt * 16 + r;
            v2f b;
            b.x = (hh == 0) ? W1[0 * 64 + col] : 0.0f;
            b.y = (hh == 0) ? W1[1 * 64 + col] : 0.0f;
            acc[nt] = wmma_f32_16x16x4(a, b, acc[nt]);
        }
    } else {
        for (int k = 0; k < 64; k += 4) {
            int c0 = k + 2 * hh;
            int base = rowc * 64 + c0;
            v2f a;
            a.x = h_in[base + 0] + agg[base + 0];
            a.y = h_in[base + 1] + agg[base + 1];
            for (int nt = 0; nt < 4; ++nt) {
                int col = nt * 16 + r;
                v2f b;
                b.x = W1[(c0 + 0) * 64 + col];
                b.y = W1[(c0 + 1) * 64 + col];
                acc[nt] = wmma_f32_16x16x4(a, b, acc[nt]);
            }
        }
    }

    // bias + relu, stage intermediate in LDS (re-layout C -> A)
    for (int nt = 0; nt < 4; ++nt) {
        int col = nt * 16 + r;
        float bias = b1[col];
        #pragma unroll
        for (int v = 0; v < 8; ++v) {
            float val = fmaxf(acc[nt][v] + bias, 0.0f);
            int m = v + 8 * hh;
            t_lds[m * 64 + col] = val;
        }
    }
    __syncthreads();

    // -------------------- matmul 2: t @ W2 --------------------
    v8f acc2[4] = {v8f{}, v8f{}, v8f{}, v8f{}};
    for (int k = 0; k < 64; k += 4) {
        int c0 = k + 2 * hh;
        v2f a;
        a.x = t_lds[r * 64 + c0 + 0];
        a.y = t_lds[r * 64 + c0 + 1];
        for (int nt = 0; nt < 4; ++nt) {
            int col = nt * 16 + r;
            v2f b;
            b.x = W2[(c0 + 0) * 64 + col];
            b.y = W2[(c0 + 1) * 64 + col];
            acc2[nt] = wmma_f32_16x16x4(a, b, acc2[nt]);
        }
    }

    for (int nt = 0; nt < 4; ++nt) {
        int col = nt * 16 + r;
        float bias = b2[col];
        #pragma unroll
        for (int v = 0; v < 8; ++v) {
            float val = fmaxf(acc2[nt][v] + bias, 0.0f);
            int m = v + 8 * hh;
            if (node0 + m < n_nodes) h_out[(node0 + m) * 64 + col] = val;
        }
    }
}

// ---------------------------------------------------------------------------
// Masked group sums (u / v / other). u/v are rare (atomics OK); "other" is
// ~all nodes -> accumulate in a register, single LDS atomic per thread.
// ---------------------------------------------------------------------------
__global__ __launch_bounds__(256) void gin_masked_sums(
    const float* __restrict__ h, const unsigned char* __restrict__ u_mask,
    const unsigned char* __restrict__ v_mask, float* __restrict__ sums,
    int n_nodes)
{
    __shared__ float ls[3 * 64];
    int t = threadIdx.x;
    if (t < 192) ls[t] = 0.0f;
    __syncthreads();

    int f = t & 63;                          // feature
    int npb = blockDim.x >> 6;               // nodes per block iteration (4)
    float other_acc = 0.0f;
    for (int n = blockIdx.x * npb + (t >> 6); n < n_nodes; n += gridDim.x * npb) {
        float val = h[n * 64 + f];
        bool us = u_mask[n] != 0;
        bool vs = v_mask[n] != 0;
        if (us) atomicAdd(&ls[0 * 64 + f], val);
        if (vs) atomicAdd(&ls[1 * 64 + f], val);
        if (!us && !vs) other_acc += val;
    }
    atomicAdd(&ls[2 * 64 + f], other_acc);
    __syncthreads();
    if (t < 192) atomicAdd(&sums[t], ls[t]);
}

__global__ __launch_bounds__(256) void gin_counts(
    const unsigned char* __restrict__ u_mask,
    const unsigned char* __restrict__ v_mask, int* __restrict__ counts, int n)
{
    int i = blockIdx.x * blockDim.x + threadIdx.x;
    if (i < n) {
        bool us = u_mask[i] != 0;
        bool vs = v_mask[i] != 0;
        if (us)         atomicAdd(&counts[0], 1);
        if (vs)         atomicAdd(&counts[1], 1);
        if (!us && !vs) atomicAdd(&counts[2], 1);
    }
}

__global__ __launch_bounds__(192) void gin_finalize(
    const float* __restrict__ sums, const int* __restrict__ counts,
    float* __restrict__ out, int layer)
{
    int t = threadIdx.x;
    if (t < 192) {
        int g = t >> 6;
        int c = counts[g];
        float denom = (float)(c > 1 ? c : 1);
        out[layer * 192 + t] = (c > 0) ? sums[t] / denom : 0.0f;
    }
}

// ---------------------------------------------------------------------------
extern "C" void kernel_launch(void* const* d_in, const int* in_sizes, int n_in,
                              void* d_out, int out_size, void* d_ws, size_t ws_size,
                              hipStream_t stream)
{
    const float*         feat = (const float*)d_in[0];
    const int*           src  = (const int*)d_in[1];
    const int*           dst  = (const int*)d_in[2];
    const unsigned char* um   = (const unsigned char*)d_in[3];
    const unsigned char* vm   = (const unsigned char*)d_in[4];
    const float* W1[3] = {(const float*)d_in[5],  (const float*)d_in[9],  (const float*)d_in[13]};
    const float* B1[3] = {(const float*)d_in[6],  (const float*)d_in[10], (const float*)d_in[14]};
    const float* W2[3] = {(const float*)d_in[7],  (const float*)d_in[11], (const float*)d_in[15]};
    const float* B2[3] = {(const float*)d_in[8],  (const float*)d_in[12], (const float*)d_in[16]};

    const int N = in_sizes[0] / 2;   // feat is [N, 2]
    const int E = in_sizes[1];

    float* ws   = (float*)d_ws;
    size_t n64  = (size_t)N * 64;
    float* hA   = ws;
    float* hB   = ws + n64;
    float* agg  = ws + 2 * n64;
    float* sums = ws + 3 * n64;
    int*   counts = (int*)(sums + 192);

    hipMemsetAsync(counts, 0, 3 * sizeof(int), stream);
    gin_counts<<<(N + 255) / 256, 256, 0, stream>>>(um, vm, counts, N);

    const float* hin = feat;
    float* bufs[2] = {hA, hB};
    for (int l = 0; l < 3; ++l) {
        int d = (l == 0) ? 2 : 64;
        float* hout = bufs[l & 1];

        hipMemsetAsync(agg, 0, (size_t)N * d * sizeof(float), stream);
        if (d == 2) {
            gin_scatter_d2<<<(E + 255) / 256, 256, 0, stream>>>(hin, src, dst, agg, E);
        } else {
            long long work = (long long)E * 64;
            int blocks = (int)((work + 255) / 256);
            gin_scatter_d64<<<blocks, 256, 0, stream>>>(hin, src, dst, agg, E);
        }

        gin_mlp_wmma<<<(N + 15) / 16, 32, 0, stream>>>(
            hin, agg, W1[l], B1[l], W2[l], B2[l], hout, N, d);

        hipMemsetAsync(sums, 0, 192 * sizeof(float), stream);
        gin_masked_sums<<<512, 256, 0, stream>>>(hout, um, vm, sums, N);
        gin_finalize<<<1, 192, 0, stream>>>(sums, counts, (float*)d_out, l);

        hin = hout;
    }
}